// GCN4Layer_PyG_996432412812
// MI455X (gfx1250) — compile-verified
//
#include <hip/hip_runtime.h>
#include <cmath>

#define F_IN 256
#define HDIM 128
#define CDIM 16

typedef float        v8f __attribute__((ext_vector_type(8)));
typedef float        v2f __attribute__((ext_vector_type(2)));
typedef unsigned int v4u __attribute__((ext_vector_type(4)));
typedef unsigned int v8u __attribute__((ext_vector_type(8)));

// -------------------- degree / symmetric norm --------------------
__global__ void k_deg_init(float* deg, int n) {
  int i = blockIdx.x * blockDim.x + threadIdx.x;
  if (i < n) deg[i] = 1.0f;  // self loop contributes 1
}
__global__ void k_deg_count(const int* __restrict__ dst, float* deg, int e) {
  int i = blockIdx.x * blockDim.x + threadIdx.x;
  if (i < e) atomicAdd(&deg[dst[i]], 1.0f);
}
__global__ void k_deg_rsqrt(float* deg, int n) {
  int i = blockIdx.x * blockDim.x + threadIdx.x;
  if (i < n) deg[i] = rsqrtf(deg[i]);
}

// -------------------- WMMA GEMM: C[nrow x F] = A[nrow x K] * B[K x F] --------------------
// One block (8 wave32) = one 16-row strip of A x all F/16 = 8 column tiles.
// A strip (16 x K f32 = 8/16 KB) is staged into LDS once by the Tensor Data Mover;
// each wave then computes one 16x16 tile with V_WMMA_F32_16X16X4_F32, reading A
// fragments from LDS (ds_load) and B fragments from L2-resident weights.
template <int K>
__global__ void __launch_bounds__(256) k_gemm_wmma(const float* __restrict__ A,
                                                   const float* __restrict__ B,
                                                   float* __restrict__ C, int nrow) {
  constexpr int F = HDIM;            // 128
  __shared__ float Alds[16 * K];

  const int tm   = blockIdx.x;       // 16-row strip index
  const int tn   = threadIdx.x >> 5; // wave -> column tile (0..7)
  const int lane = threadIdx.x & 31;
  const int m    = lane & 15;        // row in tile (A) / col in tile (B,C)
  const int g    = lane >> 4;        // half-wave group

  // ---- TDM async tile load: A[tm*16 .. tm*16+15][0..K) -> LDS (wave 0 issues) ----
  if (threadIdx.x < 32) {
    const float* gsrc = A + (size_t)tm * 16 * K;
    unsigned long long ga = (unsigned long long)(uintptr_t)gsrc;
    unsigned int laddr = (unsigned int)(uintptr_t)&Alds[0];  // addrspace(3) byte addr
    // D# group 0 (ISA 8.3): count=1 | lds_addr | global_addr[56:0] | type=2
    v4u g0;
    g0[0] = 1u;
    g0[1] = laddr;
    g0[2] = (unsigned int)ga;
    g0[3] = ((unsigned int)(ga >> 32) & 0x01FFFFFFu) | (2u << 30);
    // D# group 1 (ISA 8.4): data_size=2(4B); tensor_dim0=K; tensor_dim1=16;
    // tile_dim0=K; tile_dim1=16; tensor_dim0_stride=K
    v8u g1;
    g1[0] = (2u << 16);
    g1[1] = ((unsigned int)K & 0xFFFFu) << 16;
    g1[2] = ((unsigned int)K >> 16) | (16u << 16);
    g1[3] = ((unsigned int)K & 0xFFFFu) << 16;
    g1[4] = 16u;
    g1[5] = (unsigned int)K;
    g1[6] = 0u;
    g1[7] = 0u;
    v4u gz = {0u, 0u, 0u, 0u};       // groups 2/3: unused dims (2-D tensor)
    asm volatile("tensor_load_to_lds %0, %1, %2, %3"
                 :
                 : "s"(g0), "s"(g1), "s"(gz), "s"(gz)
                 : "memory");
    __builtin_amdgcn_s_wait_tensorcnt(0);
  }
  __syncthreads();

  if ((tm << 4) >= nrow) return;

  // A fragment base in LDS: row m, k-pair selected by g (ISA 7.12.2: 16x4 f32 layout)
  const float* Arow = &Alds[m * K + 2 * g];
  // B fragment base: rows (k+2g, k+2g+1), column tn*16 + m
  const float* Bcol = B + (size_t)(2 * g) * F + (tn << 4) + m;

  v8f acc = {0.f, 0.f, 0.f, 0.f, 0.f, 0.f, 0.f, 0.f};
#pragma unroll 8
  for (int k = 0; k < K; k += 4) {
    v2f a, b;
    a.x = Arow[k];
    a.y = Arow[k + 1];
    b.x = Bcol[k * F];
    b.y = Bcol[k * F + F];
    acc = __builtin_amdgcn_wmma_f32_16x16x4_f32(false, a, false, b, (short)0, acc,
                                                false, false);
  }

  // C/D layout: VGPR r holds row (r + 8*g), col = lane&15 (ISA 7.12.2)
  float* Cp = C + (size_t)((tm << 4) + (g << 3)) * F + (tn << 4) + m;
#pragma unroll
  for (int r = 0; r < 8; ++r) Cp[(size_t)r * F] = acc[r];
}

// -------------------- fused bias + self-loop: H = b + M * dinv^2 --------------------
__global__ void k_prep(const float* __restrict__ Mb, const float* __restrict__ bias,
                       const float* __restrict__ dinv, float* __restrict__ agg,
                       int total) {
  int i = blockIdx.x * blockDim.x + threadIdx.x;
  if (i >= total) return;
  int node = i >> 7;  // HDIM == 128
  int f = i & 127;
  float w = dinv[node];
  agg[i] = bias[f] + Mb[i] * w * w;
}

// -------------------- edge scatter: one wave per edge, float4 per lane --------------------
__global__ void k_scatter(const float* __restrict__ Mb, const int* __restrict__ src,
                          const int* __restrict__ dst, const float* __restrict__ dinv,
                          float* __restrict__ agg, int e) {
  int eidx = blockIdx.x * (blockDim.x >> 5) + (threadIdx.x >> 5);
  if (eidx >= e) return;
  int lane = threadIdx.x & 31;
  int s = src[eidx], d = dst[eidx];
  const float4* mrow = (const float4*)(Mb + (size_t)s * HDIM);
  __builtin_prefetch(&mrow[lane], 0, 0);  // global_prefetch into L2
  float w = dinv[s] * dinv[d];
  float4 v = mrow[lane];
  float* out = agg + (size_t)d * HDIM + lane * 4;
  atomicAdd(out + 0, v.x * w);
  atomicAdd(out + 1, v.y * w);
  atomicAdd(out + 2, v.z * w);
  atomicAdd(out + 3, v.w * w);
}

__global__ void k_relu(float* x, int total) {
  int i = blockIdx.x * blockDim.x + threadIdx.x;
  if (i < total) x[i] = fmaxf(x[i], 0.0f);
}

// -------------------- head: logits = h@Wl + bl, then log_softmax --------------------
__global__ void k_head(const float* __restrict__ h, const float* __restrict__ Wl,
                       const float* __restrict__ bl, float* __restrict__ out, int n) {
  int i = blockIdx.x * blockDim.x + threadIdx.x;
  if (i >= n) return;
  const float* hr = h + (size_t)i * HDIM;
  float logit[CDIM];
#pragma unroll
  for (int c = 0; c < CDIM; ++c) logit[c] = bl[c];
  for (int k = 0; k < HDIM; ++k) {
    float hv = hr[k];
    const float* w = Wl + k * CDIM;
#pragma unroll
    for (int c = 0; c < CDIM; ++c) logit[c] = fmaf(hv, w[c], logit[c]);
  }
  float mx = logit[0];
#pragma unroll
  for (int c = 1; c < CDIM; ++c) mx = fmaxf(mx, logit[c]);
  float s = 0.f;
#pragma unroll
  for (int c = 0; c < CDIM; ++c) s += expf(logit[c] - mx);
  float lse = mx + logf(s);
  float* o = out + (size_t)i * CDIM;
#pragma unroll
  for (int c = 0; c < CDIM; ++c) o[c] = logit[c] - lse;
}

extern "C" void kernel_launch(void* const* d_in, const int* in_sizes, int n_in,
                              void* d_out, int out_size, void* d_ws, size_t ws_size,
                              hipStream_t stream) {
  const float* x  = (const float*)d_in[0];
  const int*   ei = (const int*)d_in[1];
  const float* W1 = (const float*)d_in[2];
  const float* b1 = (const float*)d_in[3];
  const float* W2 = (const float*)d_in[4];
  const float* b2 = (const float*)d_in[5];
  const float* W3 = (const float*)d_in[6];
  const float* b3 = (const float*)d_in[7];
  const float* W4 = (const float*)d_in[8];
  const float* b4 = (const float*)d_in[9];
  const float* Wl = (const float*)d_in[10];
  const float* bl = (const float*)d_in[11];

  const int N = in_sizes[0] / F_IN;
  const int E = in_sizes[1] / 2;
  const int* src = ei;
  const int* dst = ei + E;

  // workspace layout: dinv[N] | M[N*HDIM] | H[N*HDIM]
  float* dinv = (float*)d_ws;
  float* Mb   = dinv + N;
  float* Hb   = Mb + (size_t)N * HDIM;

  const int total = N * HDIM;

  // 1) symmetric GCN norm
  k_deg_init<<<(N + 255) / 256, 256, 0, stream>>>(dinv, N);
  k_deg_count<<<(E + 255) / 256, 256, 0, stream>>>(dst, dinv, E);
  k_deg_rsqrt<<<(N + 255) / 256, 256, 0, stream>>>(dinv, N);

  const int rowBlocks = (N + 15) / 16;  // one 16-row strip per block

  auto post = [&](const float* b, bool relu) {
    k_prep<<<(total + 255) / 256, 256, 0, stream>>>(Mb, b, dinv, Hb, total);
    k_scatter<<<(E + 7) / 8, 256, 0, stream>>>(Mb, src, dst, dinv, Hb, E);
    if (relu) k_relu<<<(total + 255) / 256, 256, 0, stream>>>(Hb, total);
  };

  k_gemm_wmma<F_IN><<<rowBlocks, 256, 0, stream>>>(x, W1, Mb, N);
  post(b1, true);
  k_gemm_wmma<HDIM><<<rowBlocks, 256, 0, stream>>>(Hb, W2, Mb, N);
  post(b2, false);
  k_gemm_wmma<HDIM><<<rowBlocks, 256, 0, stream>>>(Hb, W3, Mb, N);
  post(b3, true);
  k_gemm_wmma<HDIM><<<rowBlocks, 256, 0, stream>>>(Hb, W4, Mb, N);
  post(b4, false);

  k_head<<<(N + 255) / 256, 256, 0, stream>>>(Hb, Wl, bl, (float*)d_out, N);
}